// TemporalGCN_19069654794550
// MI455X (gfx1250) — compile-verified
//
#include <hip/hip_runtime.h>
#include <hip/hip_bf16.h>

typedef float v2f __attribute__((ext_vector_type(2)));
typedef float v8f __attribute__((ext_vector_type(8)));

#define N_NODES 100000
#define N_EDGES 1600000
#define IN_DIM  128
#define HID_DIM 64
#define OUT_DIM 16

// ---------------------------------------------------------------------------
// Degree / normalization kernels
// ---------------------------------------------------------------------------
__global__ void k_fill_deg(float* __restrict__ deg, int n) {
    int i = blockIdx.x * blockDim.x + threadIdx.x;
    if (i < n) deg[i] = 1.0f;  // self-loop contribution
}

__global__ void k_deg_scatter(const int* __restrict__ dst, float* __restrict__ deg, int e) {
    int i = blockIdx.x * blockDim.x + threadIdx.x;
    if (i < e) atomicAdd(&deg[dst[i]], 1.0f);
}

__global__ void k_dinv(const float* __restrict__ deg, float* __restrict__ dinv, int n) {
    int i = blockIdx.x * blockDim.x + threadIdx.x;
    if (i < n) {
        float d = deg[i];
        dinv[i] = (d > 0.0f) ? rsqrtf(d) : 0.0f;
    }
}

// ---------------------------------------------------------------------------
// WMMA f32 GEMM: C[M x NC] = op(A[M x K]) * B[K x NC] (+ bias)
// One wave32 owns a 16-row strip and ALL NC columns: the A fragment is loaded
// once per k-step and reused across NC/16 back-to-back WMMAs (4x less A
// traffic than one-tile-per-wave for NC=64; B is tiny and L2-resident).
// A-frag (16x4 f32): lane m = lane&15, k-offset kk = 2*(lane>>4); 2 floats/lane.
// B-frag (4x16 f32): lane n = lane&15, rows k+kk, k+kk+1.
// C/D (16x16 f32, 8 VGPRs): row = v + 8*(lane>>4), col = lane&15.
// M % 16 == 0, K % 4 == 0, NC % 16 == 0.
// ---------------------------------------------------------------------------
template <int K, int NC, bool RELU_IN, bool ADD_BIAS>
__global__ void k_gemm_wmma(const float* __restrict__ A, const float* __restrict__ B,
                            const float* __restrict__ bias, float* __restrict__ C, int M) {
    constexpr int CT = NC / 16;                       // column tiles per wave
    const int wave = (int)((blockIdx.x * blockDim.x + threadIdx.x) >> 5);
    const int lane = (int)(threadIdx.x & 31);
    if (wave >= M / 16) return;                       // wave-uniform; EXEC stays all-1s

    const int r0 = wave * 16;
    const int m  = lane & 15;                         // also the output column-in-tile
    const int kk = (lane >> 4) << 1;                  // 0 or 2

    v8f acc[CT];
#pragma unroll
    for (int t = 0; t < CT; ++t) acc[t] = {};

    const float* __restrict__ arow = A + (size_t)(r0 + m) * K;

#pragma unroll
    for (int k = 0; k < K; k += 4) {
        float a0 = arow[k + kk];
        float a1 = arow[k + kk + 1];
        if (RELU_IN) { a0 = fmaxf(a0, 0.0f); a1 = fmaxf(a1, 0.0f); }
        v2f a; a.x = a0; a.y = a1;
        const float* __restrict__ brow0 = B + (size_t)(k + kk) * NC + m;
        const float* __restrict__ brow1 = B + (size_t)(k + kk + 1) * NC + m;
#pragma unroll
        for (int t = 0; t < CT; ++t) {
            v2f b;
            b.x = brow0[16 * t];
            b.y = brow1[16 * t];
            acc[t] = __builtin_amdgcn_wmma_f32_16x16x4_f32(
                /*neg_a=*/false, a, /*neg_b=*/false, b,
                /*c_mod=*/(short)0, acc[t], /*reuse_a=*/false, /*reuse_b=*/false);
        }
    }

    const int rbase = r0 + ((lane >> 4) << 3);
#pragma unroll
    for (int t = 0; t < CT; ++t) {
        const float bv = ADD_BIAS ? bias[16 * t + m] : 0.0f;
#pragma unroll
        for (int v = 0; v < 8; ++v) {
            C[(size_t)(rbase + v) * NC + 16 * t + m] = acc[t][v] + bv;
        }
    }
}

// ---------------------------------------------------------------------------
// agg[i][f] = b[f] + dinv[i]^2 * t[i][f]   (self-loop message + bias, no atomics)
// ---------------------------------------------------------------------------
template <int F>
__global__ void k_agg_init(const float* __restrict__ t, const float* __restrict__ dinv,
                           const float* __restrict__ b, float* __restrict__ agg, int n) {
    int i = blockIdx.x * blockDim.x + threadIdx.x;
    if (i >= n * F) return;
    int node = i / F;
    int f    = i & (F - 1);
    float di = dinv[node];
    agg[i] = b[f] + di * di * t[i];
}

// ---------------------------------------------------------------------------
// Edge scatter: agg[dst] += t[src] * (dinv[src]*dinv[dst]) across F features.
// One thread per edge; float4 row reads (rows are 256B aligned), f32 atomics.
// ---------------------------------------------------------------------------
template <int F>
__global__ void k_edge_scatter(const int* __restrict__ src, const int* __restrict__ dst,
                               const float* __restrict__ dinv, const float* __restrict__ t,
                               float* __restrict__ agg, int e) {
    int i = blockIdx.x * blockDim.x + threadIdx.x;
    if (i >= e) return;
    const int s = src[i];
    const int d = dst[i];
    const float w = dinv[s] * dinv[d];
    const float4* __restrict__ hrow = (const float4*)(t + (size_t)s * F);
    float* __restrict__ arow = agg + (size_t)d * F;
#pragma unroll
    for (int q = 0; q < F / 4; ++q) {
        float4 hv = hrow[q];
        atomicAdd(arow + 4 * q + 0, hv.x * w);
        atomicAdd(arow + 4 * q + 1, hv.y * w);
        atomicAdd(arow + 4 * q + 2, hv.z * w);
        atomicAdd(arow + 4 * q + 3, hv.w * w);
    }
}

// ---------------------------------------------------------------------------
// Host-side launch
// ---------------------------------------------------------------------------
extern "C" void kernel_launch(void* const* d_in, const int* in_sizes, int n_in,
                              void* d_out, int out_size, void* d_ws, size_t ws_size,
                              hipStream_t stream) {
    (void)in_sizes; (void)n_in; (void)out_size; (void)ws_size;

    const float* x  = (const float*)d_in[0];
    const int*   ei = (const int*)d_in[1];
    const float* W1 = (const float*)d_in[2];
    const float* b1 = (const float*)d_in[3];
    const float* W2 = (const float*)d_in[4];
    const float* b2 = (const float*)d_in[5];
    const float* Wc = (const float*)d_in[6];
    const float* bc = (const float*)d_in[7];
    float* out = (float*)d_out;

    const int* src = ei;            // edge_index[0]
    const int* dst = ei + N_EDGES;  // edge_index[1]

    // Workspace layout (256B-aligned offsets)
    char* ws = (char*)d_ws;
    float* dinv = (float*)(ws + 0);          //   400,000 B
    float* deg  = (float*)(ws + 400384);     //   400,000 B
    float* t    = (float*)(ws + 800768);     // 25,600,000 B (reused for both layers)
    float* agg1 = (float*)(ws + 26400768);   // 25,600,000 B
    float* agg2 = (float*)(ws + 52000768);   // 25,600,000 B -> total ~77.6 MB

    const dim3 blk(256);
    const int gN  = (N_NODES + 255) / 256;
    const int gE  = (N_EDGES + 255) / 256;
    const int gNF = (N_NODES * HID_DIM + 255) / 256;
    const int gG  = ((N_NODES / 16) + 7) / 8;   // 782 blocks, 8 row-strip waves each

    // gcn_norm
    k_fill_deg<<<gN, blk, 0, stream>>>(deg, N_NODES);
    k_deg_scatter<<<gE, blk, 0, stream>>>(dst, deg, N_EDGES);
    k_dinv<<<gN, blk, 0, stream>>>(deg, dinv, N_NODES);

    // Layer 1: t = x @ W1 ; agg1 = b1 + scatter(norm * t)
    k_gemm_wmma<IN_DIM, HID_DIM, false, false><<<gG, blk, 0, stream>>>(x, W1, nullptr, t, N_NODES);
    k_agg_init<HID_DIM><<<gNF, blk, 0, stream>>>(t, dinv, b1, agg1, N_NODES);
    k_edge_scatter<HID_DIM><<<gE, blk, 0, stream>>>(src, dst, dinv, t, agg1, N_EDGES);

    // Layer 2: t = relu(agg1) @ W2 ; agg2 = b2 + scatter(norm * t)
    k_gemm_wmma<HID_DIM, HID_DIM, true, false><<<gG, blk, 0, stream>>>(agg1, W2, nullptr, t, N_NODES);
    k_agg_init<HID_DIM><<<gNF, blk, 0, stream>>>(t, dinv, b2, agg2, N_NODES);
    k_edge_scatter<HID_DIM><<<gE, blk, 0, stream>>>(src, dst, dinv, t, agg2, N_EDGES);

    // Classifier: out = agg2 @ Wc + bc
    k_gemm_wmma<HID_DIM, OUT_DIM, false, true><<<gG, blk, 0, stream>>>(agg2, Wc, bc, out, N_NODES);
}